// GroupChatAModel_Adr_18803366822318
// MI455X (gfx1250) — compile-verified
//
#include <hip/hip_runtime.h>
#include <math.h>

// CDNA5 / gfx1250: wave32, WMMA fp32 16x16x4 tiles.
typedef __attribute__((ext_vector_type(2))) float v2f;
typedef __attribute__((ext_vector_type(8))) float v8f;

// ---------------------------------------------------------------------------
// One K-pass of a 16x16 output tile:  C += A[16xK] * W[ncol0..ncol0+15, 0..K)^T
// A is row-major with leading dim lda (LDS or global).
// W is row-major [Ngates, K] (global, L2-resident): B[k][n] = W[(ncol0+n)*ldw + k].
// Lane layout per CDNA5 ISA (V_WMMA_F32_16X16X4_F32):
//   A: lane<16 holds row (lane), K pair {k,k+1}; lane>=16 row (lane-16), {k+2,k+3}
//   B: lane<16 holds col (lane), K rows {k,k+1}; lane>=16 col (lane-16), {k+2,k+3}
//   C: VGPR v, lane<16 -> (M=v, N=lane); lane>=16 -> (M=v+8, N=lane-16)
// ---------------------------------------------------------------------------
__device__ __forceinline__ v8f gemm_pass(const float* __restrict__ A, int lda,
                                         const float* __restrict__ W, int ldw,
                                         int ncol0, int K, v8f c, int lane) {
  const int r16 = lane & 15;
  const int kh  = lane >> 4;
  const float* a = A + r16 * lda + 2 * kh;
  const float* w = W + (ncol0 + r16) * ldw + 2 * kh;
  for (int kk = 0; kk < K; kk += 4) {
    v2f av = *(const v2f*)(a + kk);
    v2f bv = *(const v2f*)(w + kk);
    c = __builtin_amdgcn_wmma_f32_16x16x4_f32(false, av, false, bv,
                                              (short)0, c, false, false);
  }
  return c;
}

__device__ __forceinline__ void store_tile(float* __restrict__ dst, int ld,
                                           int m0, int j0, v8f c, int lane) {
  const int r16 = lane & 15;
  const int kh  = lane >> 4;
#pragma unroll
  for (int v = 0; v < 8; ++v)
    dst[(m0 + v + 8 * kh) * ld + j0 + r16] = c[v];
}

__device__ __forceinline__ float sigmoidf_(float x) {
  return 1.0f / (1.0f + __expf(-x));
}

// ---------------------------------------------------------------------------
// Fused token-GRU encoder (utterance & response encoders).
// Each workgroup (256 thr = 8 waves) owns 16 sequences; h lives in LDS across
// all T steps; per step: gather embeddings -> WMMA gates in registers -> cell.
// Sequence n maps to (x = n % divN, b = n / divN); out_last is [x, Bn, H];
// out_all (optional) is [x, T, Bn, H].
// ---------------------------------------------------------------------------
__global__ __launch_bounds__(256) void gru_seq_kernel(
    const int* __restrict__ tokens,      // [N, T]
    const float* __restrict__ emb,       // [V, 256]
    const float* __restrict__ Wih,       // [768, 256]
    const float* __restrict__ Whh,       // [768, 256]
    const float* __restrict__ bih,       // [768]
    const float* __restrict__ bhh,       // [768]
    float* __restrict__ out_last,
    float* __restrict__ out_all,
    int T, int divN, int Bn) {
  __shared__ float xs[16 * 260];   // padded stride: bank-conflict-free A reads
  __shared__ float hb[16 * 260];
  const int tid  = threadIdx.x;
  const int lane = tid & 31;
  const int wave = tid >> 5;
  const int r16  = lane & 15;
  const int kh   = lane >> 4;
  const int n0   = blockIdx.x * 16;

  for (int e = tid; e < 16 * 260; e += 256) hb[e] = 0.0f;
  __syncthreads();

  for (int t = 0; t < T; ++t) {
    // gather x_t: 16 rows x 256 f32 (each thread: one 16-float row segment)
    {
      const int m = tid >> 4, seg = tid & 15;
      const int tok = tokens[(n0 + m) * T + t];
      const float4* src = (const float4*)(emb + (long)tok * 256 + seg * 16);
      float4* dst = (float4*)(xs + m * 260 + seg * 16);
      dst[0] = src[0]; dst[1] = src[1]; dst[2] = src[2]; dst[3] = src[3];
    }
    __syncthreads();

    // gate GEMMs: 16 column groups of 16, 2 per wave, kept in registers
    v8f Cr[2], Cz[2], Cni[2], Cnh[2];
#pragma unroll
    for (int gg = 0; gg < 2; ++gg) {
      const int j0 = (wave + gg * 8) * 16;
      {
        v8f c = {};
        c = gemm_pass(xs, 260, Wih, 256, j0, 256, c, lane);
        c = gemm_pass(hb, 260, Whh, 256, j0, 256, c, lane);
        Cr[gg] = c;
      }
      {
        v8f c = {};
        c = gemm_pass(xs, 260, Wih, 256, j0 + 256, 256, c, lane);
        c = gemm_pass(hb, 260, Whh, 256, j0 + 256, 256, c, lane);
        Cz[gg] = c;
      }
      {
        v8f c = {};
        c = gemm_pass(xs, 260, Wih, 256, j0 + 512, 256, c, lane);
        Cni[gg] = c;
      }
      {
        v8f c = {};
        c = gemm_pass(hb, 260, Whh, 256, j0 + 512, 256, c, lane);
        Cnh[gg] = c;
      }
    }
    __syncthreads();  // all reads of hb complete before updates

    // GRU cell directly on WMMA accumulators
#pragma unroll
    for (int gg = 0; gg < 2; ++gg) {
      const int j0 = (wave + gg * 8) * 16;
      const int j  = j0 + r16;
      const float bri = bih[j],       brh = bhh[j];
      const float bzi = bih[j + 256], bzh = bhh[j + 256];
      const float bni = bih[j + 512], bnh = bhh[j + 512];
#pragma unroll
      for (int v = 0; v < 8; ++v) {
        const int m = v + 8 * kh;
        const float hold = hb[m * 260 + j];
        const float r  = sigmoidf_(Cr[gg][v] + bri + brh);
        const float z  = sigmoidf_(Cz[gg][v] + bzi + bzh);
        const float nn = tanhf(Cni[gg][v] + bni + r * (Cnh[gg][v] + bnh));
        const float hnew = (1.0f - z) * nn + z * hold;
        hb[m * 260 + j] = hnew;
        if (out_all) {
          const int n = n0 + m;
          const int x = n % divN, bb = n / divN;
          out_all[(((long)x * T + t) * Bn + bb) * 256 + j] = hnew;
        }
      }
    }
    __syncthreads();
  }

  // final hidden state
  {
    const int m = tid >> 4, seg = tid & 15;
    const int n = n0 + m;
    const int x = n % divN, bb = n / divN;
    float4* dst = (float4*)(out_last + ((long)x * Bn + bb) * 256 + seg * 16);
    const float4* src = (const float4*)(hb + m * 260 + seg * 16);
    dst[0] = src[0]; dst[1] = src[1]; dst[2] = src[2]; dst[3] = src[3];
  }
}

// ---------------------------------------------------------------------------
// Context GRU: one persistent workgroup (512 thr = 16 waves), 64 rows, 50
// steps. Input-gate GEMM fused (x = ctx_in[s], K=512, read from global/L2);
// h in LDS. Gate tiles staged in a small global scratch G (4 planes 64x256).
// ---------------------------------------------------------------------------
__global__ __launch_bounds__(512) void ctx_gru_kernel(
    const float* __restrict__ xin,   // [50*64, 512]
    const float* __restrict__ Wih,   // [768, 512]
    const float* __restrict__ Whh,   // [768, 256]
    const float* __restrict__ bih, const float* __restrict__ bhh,
    float* __restrict__ ctx_out,     // [50*64, 256]
    float* __restrict__ ctx_hid,     // [64, 256]
    float* __restrict__ G) {
  __shared__ float hb[64 * 256];
  const int tid = threadIdx.x, lane = tid & 31, wave = tid >> 5;
  float* Gr  = G;
  float* Gz  = G + 16384;
  float* Gni = G + 32768;
  float* Gnh = G + 49152;

  for (int e = tid; e < 64 * 256; e += 512) hb[e] = 0.0f;
  __syncthreads();

  for (int s = 0; s < 50; ++s) {
    const float* xs_ = xin + (long)s * 64 * 512;
    for (int u = wave; u < 64; u += 16) {
      const int m0 = (u & 3) * 16, j0 = (u >> 2) * 16;
      {
        v8f c = {};
        c = gemm_pass(xs_ + m0 * 512, 512, Wih, 512, j0, 512, c, lane);
        c = gemm_pass(hb + m0 * 256, 256, Whh, 256, j0, 256, c, lane);
        store_tile(Gr, 256, m0, j0, c, lane);
      }
      {
        v8f c = {};
        c = gemm_pass(xs_ + m0 * 512, 512, Wih, 512, j0 + 256, 512, c, lane);
        c = gemm_pass(hb + m0 * 256, 256, Whh, 256, j0 + 256, 256, c, lane);
        store_tile(Gz, 256, m0, j0, c, lane);
      }
      {
        v8f c = {};
        c = gemm_pass(xs_ + m0 * 512, 512, Wih, 512, j0 + 512, 512, c, lane);
        store_tile(Gni, 256, m0, j0, c, lane);
      }
      {
        v8f c = {};
        c = gemm_pass(hb + m0 * 256, 256, Whh, 256, j0 + 512, 256, c, lane);
        store_tile(Gnh, 256, m0, j0, c, lane);
      }
    }
    __threadfence();
    __syncthreads();
    for (int e = tid; e < 64 * 256; e += 512) {
      const int j = e & 255;
      const float hold = hb[e];
      const float r  = sigmoidf_(Gr[e] + bih[j] + bhh[j]);
      const float z  = sigmoidf_(Gz[e] + bih[j + 256] + bhh[j + 256]);
      const float nn = tanhf(Gni[e] + bih[j + 512] + r * (Gnh[e] + bhh[j + 512]));
      const float hnew = (1.0f - z) * nn + z * hold;
      hb[e] = hnew;
      ctx_out[(long)s * 64 * 256 + e] = hnew;
    }
    __threadfence();
    __syncthreads();
  }
  for (int e = tid; e < 64 * 256; e += 512) ctx_hid[e] = hb[e];
}

// ---------------------------------------------------------------------------
// Speaker GRU. Only one slot per (b,s) is active each step, so we update just
// the 64 active rows (b, a=spk_agents[b][s]) -- identical to the reference's
// masked where(). gi fused: x = context_output[s] (K=256). h_spk in global ws.
// ---------------------------------------------------------------------------
__global__ __launch_bounds__(512) void spk_gru_kernel(
    const float* __restrict__ ctx_out,  // [50*64, 256]
    const int* __restrict__ agents,     // [64, 50]
    const float* __restrict__ Wih,      // [768, 256]
    const float* __restrict__ Whh,      // [768, 256]
    const float* __restrict__ bih, const float* __restrict__ bhh,
    float* __restrict__ h_spk,          // [64*51, 256], pre-zeroed
    float* __restrict__ G) {
  __shared__ float hb[64 * 256];  // gathered active rows
  const int tid = threadIdx.x, lane = tid & 31, wave = tid >> 5;
  float* Gr  = G;
  float* Gz  = G + 16384;
  float* Gni = G + 32768;
  float* Gnh = G + 49152;

  for (int s = 0; s < 50; ++s) {
    for (int e = tid; e < 64 * 256; e += 512) {
      const int b = e >> 8, j = e & 255;
      const int a = agents[b * 50 + s];
      hb[e] = h_spk[((long)b * 51 + a) * 256 + j];
    }
    __syncthreads();

    const float* xs_ = ctx_out + (long)s * 64 * 256;
    for (int u = wave; u < 64; u += 16) {
      const int m0 = (u & 3) * 16, j0 = (u >> 2) * 16;
      {
        v8f c = {};
        c = gemm_pass(xs_ + m0 * 256, 256, Wih, 256, j0, 256, c, lane);
        c = gemm_pass(hb + m0 * 256, 256, Whh, 256, j0, 256, c, lane);
        store_tile(Gr, 256, m0, j0, c, lane);
      }
      {
        v8f c = {};
        c = gemm_pass(xs_ + m0 * 256, 256, Wih, 256, j0 + 256, 256, c, lane);
        c = gemm_pass(hb + m0 * 256, 256, Whh, 256, j0 + 256, 256, c, lane);
        store_tile(Gz, 256, m0, j0, c, lane);
      }
      {
        v8f c = {};
        c = gemm_pass(xs_ + m0 * 256, 256, Wih, 256, j0 + 512, 256, c, lane);
        store_tile(Gni, 256, m0, j0, c, lane);
      }
      {
        v8f c = {};
        c = gemm_pass(hb + m0 * 256, 256, Whh, 256, j0 + 512, 256, c, lane);
        store_tile(Gnh, 256, m0, j0, c, lane);
      }
    }
    __threadfence();
    __syncthreads();

    for (int e = tid; e < 64 * 256; e += 512) {
      const int b = e >> 8, j = e & 255;
      const int a = agents[b * 50 + s];
      const float hold = hb[e];
      const float r  = sigmoidf_(Gr[e] + bih[j] + bhh[j]);
      const float z  = sigmoidf_(Gz[e] + bih[j + 256] + bhh[j + 256]);
      const float nn = tanhf(Gni[e] + bih[j + 512] + r * (Gnh[e] + bhh[j + 512]));
      const float hnew = (1.0f - z) * nn + z * hold;
      h_spk[((long)b * 51 + a) * 256 + j] = hnew;
    }
    __threadfence();
    __syncthreads();
  }
}

// ---------------------------------------------------------------------------
// Replicate np.random.seed(1) agent tables (MT19937, legacy rk_double).
// Host-side constant in the reference; regenerated deterministically per call.
// ---------------------------------------------------------------------------
__global__ void mt_agent_tables(float* __restrict__ out /* 64*51*256 */) {
  if (threadIdx.x != 0 || blockIdx.x != 0) return;
  unsigned mt[624];
  mt[0] = 1u;
  for (int i = 1; i < 624; ++i)
    mt[i] = 1812433253u * (mt[i - 1] ^ (mt[i - 1] >> 30)) + (unsigned)i;
  int mti = 624;
  auto next = [&]() -> unsigned {
    if (mti >= 624) {
      for (int k = 0; k < 624; ++k) {
        unsigned y = (mt[k] & 0x80000000u) | (mt[(k + 1) % 624] & 0x7fffffffu);
        mt[k] = mt[(k + 397) % 624] ^ (y >> 1) ^ ((y & 1u) ? 0x9908b0dfu : 0u);
      }
      mti = 0;
    }
    unsigned y = mt[mti++];
    y ^= y >> 11;
    y ^= (y << 7) & 0x9d2c5680u;
    y ^= (y << 15) & 0xefc60000u;
    y ^= y >> 18;
    return y;
  };
  const int n = 64 * 51 * 256;
  for (int i = 0; i < n; ++i) {
    const unsigned a = next() >> 5;
    const unsigned b = next() >> 6;
    const double d = ((double)a * 67108864.0 + (double)b) *
                     (1.0 / 9007199254740992.0);
    out[i] = (float)d;
  }
}

// ctx_in[s,b,:] = concat(utter_hidden[s,b,:], agent_table[b, agents[b,s], :])
__global__ void build_ctx_in(const float* __restrict__ uh,
                             const float* __restrict__ tables,
                             const int* __restrict__ agents,
                             float* __restrict__ ctx_in) {
  const long total = 50L * 64 * 512;
  for (long e = blockIdx.x * (long)blockDim.x + threadIdx.x; e < total;
       e += (long)gridDim.x * blockDim.x) {
    const long row = e >> 9;
    const int c = (int)(e & 511);
    const int s = (int)(row / 64), b = (int)(row % 64);
    float v;
    if (c < 256) {
      v = uh[row * 256 + c];
    } else {
      const int a = agents[b * 50 + s];
      v = tables[((long)b * 51 + a) * 256 + (c - 256)];
    }
    ctx_in[e] = v;
  }
}

__global__ void zero_f32(float* __restrict__ p, long n) {
  for (long i = blockIdx.x * (long)blockDim.x + threadIdx.x; i < n;
       i += (long)gridDim.x * blockDim.x)
    p[i] = 0.0f;
}

__global__ __launch_bounds__(256) void spk_finalize(
    const int* __restrict__ agents, const float* __restrict__ h_spk,
    float* __restrict__ spk_emb, float* __restrict__ mask) {
  const int p = blockIdx.x;  // 0..64*51-1
  const int b = p / 51, a = p % 51;
  bool present = false;
  for (int s = 0; s < 50; ++s) present = present || (agents[b * 50 + s] == a);
  const int j = threadIdx.x;
  const float v = present ? h_spk[((long)b * 51 + a) * 256 + j] : 0.0f;
  spk_emb[((long)b * 51 + a) * 256 + j] = v;
  if (j == 0) mask[p] = (present && a > 0) ? 1.0f : 0.0f;
}

// ---------------------------------------------------------------------------
extern "C" void kernel_launch(void* const* d_in, const int* in_sizes, int n_in,
                              void* d_out, int out_size, void* d_ws,
                              size_t ws_size, hipStream_t stream) {
  (void)in_sizes; (void)n_in; (void)out_size; (void)ws_size;
  const int*   context    = (const int*)d_in[0];
  const int*   response   = (const int*)d_in[1];
  const int*   spk_agents = (const int*)d_in[2];
  const float* emb_utter  = (const float*)d_in[3];
  const float* emb_resp   = (const float*)d_in[4];
  const float* uWih = (const float*)d_in[5];
  const float* uWhh = (const float*)d_in[6];
  const float* ubih = (const float*)d_in[7];
  const float* ubhh = (const float*)d_in[8];
  const float* cWih = (const float*)d_in[9];
  const float* cWhh = (const float*)d_in[10];
  const float* cbih = (const float*)d_in[11];
  const float* cbhh = (const float*)d_in[12];
  const float* rWih = (const float*)d_in[13];
  const float* rWhh = (const float*)d_in[14];
  const float* rbih = (const float*)d_in[15];
  const float* rbhh = (const float*)d_in[16];
  const float* sWih = (const float*)d_in[17];
  const float* sWhh = (const float*)d_in[18];
  const float* sbih = (const float*)d_in[19];
  const float* sbhh = (const float*)d_in[20];

  float* out = (float*)d_out;
  float* ctx_out  = out;                // [50,64,256]   819200
  float* ctx_hid  = out + 819200;       // [1,64,256]    16384
  float* resp_out = out + 835584;       // [2,50,64,256] 1638400
  float* resp_hid = out + 2473984;      // [2,64,256]    32768
  float* spk_emb  = out + 2506752;      // [64,51,256]   835584
  float* spk_mask = out + 3342336;      // [64,51]       3264

  float* ws = (float*)d_ws;
  float* ws_utter = ws;                 // [50,64,256]   819200
  float* ws_tab   = ws + 819200;        // [64,51,256]   835584
  float* ws_ctxin = ws + 1654784;       // [50,64,512]   1638400
  float* ws_hspk  = ws + 3293184;       // [64,51,256]   835584
  float* ws_G     = ws + 4128768;       // 4 planes of 64*256 = 65536

  mt_agent_tables<<<1, 32, 0, stream>>>(ws_tab);

  // utterance encoder: 3200 sequences, final hidden -> ws_utter [S,B,H]
  gru_seq_kernel<<<200, 256, 0, stream>>>(context, emb_utter, uWih, uWhh, ubih,
                                          ubhh, ws_utter, nullptr,
                                          50, 50, 64);
  // response encoder: 128 sequences, per-step -> resp_out, final -> resp_hid
  gru_seq_kernel<<<8, 256, 0, stream>>>(response, emb_resp, rWih, rWhh, rbih,
                                        rbhh, resp_hid, resp_out,
                                        50, 2, 64);

  build_ctx_in<<<2048, 256, 0, stream>>>(ws_utter, ws_tab, spk_agents, ws_ctxin);
  zero_f32<<<1024, 256, 0, stream>>>(ws_hspk, 64L * 51 * 256);

  ctx_gru_kernel<<<1, 512, 0, stream>>>(ws_ctxin, cWih, cWhh, cbih, cbhh,
                                        ctx_out, ctx_hid, ws_G);
  spk_gru_kernel<<<1, 512, 0, stream>>>(ctx_out, spk_agents, sWih, sWhh, sbih,
                                        sbhh, ws_hspk, ws_G);
  spk_finalize<<<64 * 51, 256, 0, stream>>>(spk_agents, ws_hspk, spk_emb,
                                            spk_mask);
}